// GraphAttentionLayer_4526895530308
// MI455X (gfx1250) — compile-verified
//
#include <hip/hip_runtime.h>
#include <math.h>

typedef __attribute__((ext_vector_type(2))) float v2f;
typedef __attribute__((ext_vector_type(8))) float v8f;

#define N_NODES 50000
#define N_EDGES 800000
#define IN_CH   128
#define HEADS   4
#define OUT_CH  32
#define FEAT_CH 128   // HEADS*OUT_CH
#define NEG_SLOPE 0.2f

// ---------------------------------------------------------------------------
// feat = x @ W  via v_wmma_f32_16x16x4_f32 (full fp32 precision).
// One wave per 16x16 output tile; block of 256 = 8 waves covers the 8 n-tiles
// of one 16-row m-tile.  A-frag layout (16x4 f32): lanes 0-15 hold M=lane,
// VGPR v holds K = 2*half + v (half = lane/16).  B-frag mirrors it on K.
// C/D: VGPR r -> row M = r + 8*half, col N = lane%16.
// ---------------------------------------------------------------------------
__global__ __launch_bounds__(256) void gat_gemm(const float* __restrict__ x,
                                                const float* __restrict__ W,
                                                float* __restrict__ feat) {
  const int wave = threadIdx.x >> 5;   // 0..7 -> n tile
  const int lane = threadIdx.x & 31;
  const int hl   = lane >> 4;          // which half of the wave
  const int lrow = lane & 15;
  const int m0 = blockIdx.x * 16;
  const int n0 = wave * 16;

  const float* __restrict__ arow = x + (size_t)(m0 + lrow) * IN_CH + 2 * hl;
  const float* __restrict__ bcol = W + n0 + lrow;

  v8f c = {};
#pragma unroll
  for (int k0 = 0; k0 < IN_CH; k0 += 4) {
    v2f a = *(const v2f*)(arow + k0);                 // A[lrow][k0+2*hl .. +1]
    v2f b;
    b.x = bcol[(size_t)(k0 + 2 * hl + 0) * FEAT_CH];  // W[k][n0+lrow]
    b.y = bcol[(size_t)(k0 + 2 * hl + 1) * FEAT_CH];
    c = __builtin_amdgcn_wmma_f32_16x16x4_f32(false, a, false, b,
                                              (short)0, c, false, false);
  }

  float* __restrict__ orow = feat + (size_t)(m0 + 8 * hl) * FEAT_CH + n0 + lrow;
#pragma unroll
  for (int r = 0; r < 8; ++r) orow[(size_t)r * FEAT_CH] = c[r];
}

// ---------------------------------------------------------------------------
// el[n,h] = feat[n,h,:] . attn_l[h,:],  er likewise.  One thread per (n,h).
// ---------------------------------------------------------------------------
__global__ void gat_node_attn(const float* __restrict__ feat,
                              const float* __restrict__ attn_l,
                              const float* __restrict__ attn_r,
                              float* __restrict__ el, float* __restrict__ er) {
  int idx = blockIdx.x * blockDim.x + threadIdx.x;   // n*HEADS + h
  if (idx >= N_NODES * HEADS) return;
  int h = idx & (HEADS - 1);
  const float* f  = feat + (size_t)(idx >> 2) * FEAT_CH + h * OUT_CH;
  const float* al = attn_l + h * OUT_CH;
  const float* ar = attn_r + h * OUT_CH;
  float sl = 0.f, sr = 0.f;
#pragma unroll
  for (int d = 0; d < OUT_CH; ++d) { float v = f[d]; sl += v * al[d]; sr += v * ar[d]; }
  el[idx] = sl;
  er[idx] = sr;
}

// ---------------------------------------------------------------------------
// Init: segment-max buffer to -inf, segment-sum to 0, out to mean-over-heads
// of bias (d_out/d_ws are poisoned by the harness, so we must clear them).
// ---------------------------------------------------------------------------
__global__ void gat_init(float* __restrict__ out, float* __restrict__ m,
                         float* __restrict__ s, const float* __restrict__ bias) {
  int idx = blockIdx.x * blockDim.x + threadIdx.x;
  if (idx < N_NODES * OUT_CH) {
    int d = idx & (OUT_CH - 1);
    out[idx] = 0.25f * (bias[d] + bias[OUT_CH + d] + bias[2 * OUT_CH + d] +
                        bias[3 * OUT_CH + d]);
  }
  if (idx < N_NODES * HEADS) {
    ((unsigned int*)m)[idx] = 0xFF800000u;   // -inf
    s[idx] = 0.f;
  }
}

// ---------------------------------------------------------------------------
// Pass 1: e = leaky_relu(el[src]+er[dst]); store e; atomic float-max into
// m[dst] via the int-max / uint-min bit trick.
// ---------------------------------------------------------------------------
__global__ void gat_edge_logits(const int* __restrict__ src,
                                const int* __restrict__ dst,
                                const float* __restrict__ el,
                                const float* __restrict__ er,
                                float* __restrict__ e_ws,
                                float* __restrict__ m) {
  int e = blockIdx.x * blockDim.x + threadIdx.x;
  if (e >= N_EDGES) return;
  int sN = src[e], dN = dst[e];
#pragma unroll
  for (int h = 0; h < HEADS; ++h) {
    float v = el[sN * HEADS + h] + er[dN * HEADS + h];
    v = v > 0.f ? v : NEG_SLOPE * v;
    e_ws[(size_t)e * HEADS + h] = v;
    if (v >= 0.f)
      atomicMax((int*)(m + dN * HEADS + h), __float_as_int(v));
    else
      atomicMin((unsigned int*)(m + dN * HEADS + h), __float_as_uint(v));
  }
}

// ---------------------------------------------------------------------------
// Pass 2: ex = exp(e - m[dst]); overwrite e_ws; atomicAdd into s[dst].
// ---------------------------------------------------------------------------
__global__ void gat_edge_exp(const int* __restrict__ dst,
                             float* __restrict__ e_ws,
                             const float* __restrict__ m,
                             float* __restrict__ s) {
  int e = blockIdx.x * blockDim.x + threadIdx.x;
  if (e >= N_EDGES) return;
  int dN = dst[e];
#pragma unroll
  for (int h = 0; h < HEADS; ++h) {
    float ex = expf(e_ws[(size_t)e * HEADS + h] - m[dN * HEADS + h]);
    e_ws[(size_t)e * HEADS + h] = ex;
    atomicAdd(s + dN * HEADS + h, ex);
  }
}

// ---------------------------------------------------------------------------
// Pass 3: one wave per edge, lane = output channel d (0..31).
// alpha_h = ex/s computed by lanes 0..3, broadcast via shfl; heads combined
// locally so only ONE atomicAdd per (edge, channel).  Head-mean folded in.
// ---------------------------------------------------------------------------
__global__ __launch_bounds__(256) void gat_aggregate(const int* __restrict__ src,
                                                     const int* __restrict__ dst,
                                                     const float* __restrict__ e_ws,
                                                     const float* __restrict__ s,
                                                     const float* __restrict__ feat,
                                                     float* __restrict__ out) {
  int lane = threadIdx.x & 31;
  int edge = blockIdx.x * (blockDim.x >> 5) + (threadIdx.x >> 5);
  if (edge >= N_EDGES) return;
  int sN = src[edge], dN = dst[edge];

  float alpha = 0.f;
  if (lane < HEADS)
    alpha = e_ws[(size_t)edge * HEADS + lane] / s[dN * HEADS + lane];
  float a0 = __shfl(alpha, 0);
  float a1 = __shfl(alpha, 1);
  float a2 = __shfl(alpha, 2);
  float a3 = __shfl(alpha, 3);

  const float* f = feat + (size_t)sN * FEAT_CH + lane;
  float acc = 0.25f * (a0 * f[0] + a1 * f[OUT_CH] + a2 * f[2 * OUT_CH] +
                       a3 * f[3 * OUT_CH]);
  atomicAdd(out + (size_t)dN * OUT_CH + lane, acc);
}

// ---------------------------------------------------------------------------
extern "C" void kernel_launch(void* const* d_in, const int* in_sizes, int n_in,
                              void* d_out, int out_size, void* d_ws, size_t ws_size,
                              hipStream_t stream) {
  const float* x      = (const float*)d_in[0];
  const int*   src    = (const int*)d_in[1];
  const int*   dst    = (const int*)d_in[2];
  const float* W      = (const float*)d_in[3];
  const float* attn_l = (const float*)d_in[4];
  const float* attn_r = (const float*)d_in[5];
  const float* bias   = (const float*)d_in[6];
  float* out = (float*)d_out;

  // workspace layout (floats)
  float* feat = (float*)d_ws;                          // N*128
  float* el   = feat + (size_t)N_NODES * FEAT_CH;      // N*4
  float* er   = el + N_NODES * HEADS;                  // N*4
  float* m    = er + N_NODES * HEADS;                  // N*4
  float* ssum = m + N_NODES * HEADS;                   // N*4
  float* e_ws = ssum + N_NODES * HEADS;                // E*4

  gat_gemm<<<N_NODES / 16, 256, 0, stream>>>(x, W, feat);
  gat_node_attn<<<(N_NODES * HEADS + 255) / 256, 256, 0, stream>>>(feat, attn_l,
                                                                   attn_r, el, er);
  gat_init<<<(N_NODES * OUT_CH + 255) / 256, 256, 0, stream>>>(out, m, ssum, bias);
  gat_edge_logits<<<(N_EDGES + 255) / 256, 256, 0, stream>>>(src, dst, el, er,
                                                             e_ws, m);
  gat_edge_exp<<<(N_EDGES + 255) / 256, 256, 0, stream>>>(dst, e_ws, m, ssum);
  gat_aggregate<<<(N_EDGES + 7) / 8, 256, 0, stream>>>(src, dst, e_ws, ssum,
                                                       feat, out);
}